// FormulaFeatureGenerator_18966575579212
// MI455X (gfx1250) — compile-verified
//
#include <hip/hip_runtime.h>

// out = exp( log(max(X,1e-5)) @ E^T ),  X:[B,4] f32, E:[625,4] f32, out:[B,625] f32
//
// Implemented as V_WMMA_F32_16X16X4_F32 tiles (A = 16x4 logX tile, B = 4x16 E^T tile),
// computed in base-2 (v_log_f32 / v_exp_f32 are base-2 on AMD, and
// exp2(log2(x) dot E) == exp(ln(x) dot E) exactly in the algebra).
// Store-bandwidth-bound: 655 MB output -> NT streaming stores.

typedef __attribute__((ext_vector_type(2))) float v2f;
typedef __attribute__((ext_vector_type(8))) float v8f;

#define BATCH   262144
#define ENC_DIM 4
#define NFEAT   625
#define NTILES  40          // ceil(625 / 16)
#define WAVES_PER_BLOCK 8   // 256 threads, wave32

__global__ __launch_bounds__(256)
void formula_feature_wmma_kernel(const float* __restrict__ X,
                                 const float* __restrict__ E,
                                 float* __restrict__ out)
{
    const int lane  = threadIdx.x & 31;
    const int wave  = threadIdx.x >> 5;
    const int mtile = blockIdx.x * WAVES_PER_BLOCK + wave;   // 16 rows per wave

    const int halfsel = lane >> 4;          // 0: lanes 0-15, 1: lanes 16-31
    const int khalf   = halfsel * 2;        // K components 0,1 or 2,3
    const int sub     = lane & 15;

    // ---- A operand: 16x4 f32 matrix (ISA layout):
    //   lanes 0-15  : VGPR0 = K0, VGPR1 = K1, row M = lane
    //   lanes 16-31 : VGPR0 = K2, VGPR1 = K3, row M = lane-16
    const int arow = mtile * 16 + sub;
    const float2 xv = *reinterpret_cast<const float2*>(X + (size_t)arow * ENC_DIM + khalf);
    v2f a;
    a.x = __builtin_amdgcn_logf(fmaxf(xv.x, 1e-5f));   // v_log_f32 (log2)
    a.y = __builtin_amdgcn_logf(fmaxf(xv.y, 1e-5f));

    const int mbase = mtile * 16 + halfsel * 8;   // C/D row base for this half-wave

    for (int t = 0; t < NTILES; ++t) {
        const int col = t * 16 + sub;
        // ---- B operand: 4x16 f32 tile of E^T (mirror layout of A):
        //   lanes 0-15  : VGPR0 = K0, VGPR1 = K1, col N = sub
        //   lanes 16-31 : VGPR0 = K2, VGPR1 = K3
        // Clamp the tail column so every lane issues a valid load (EXEC must stay all-1s
        // for the WMMA); tail results are simply not stored.
        const int ncl = col < NFEAT ? col : (NFEAT - 1);
        const float2 ev = *reinterpret_cast<const float2*>(E + (size_t)ncl * ENC_DIM + khalf);
        v2f b;
        b.x = ev.x;
        b.y = ev.y;

        v8f c = {};
        // 8 args: (neg_a, A, neg_b, B, c_mod, C, reuse_a, reuse_b)
        c = __builtin_amdgcn_wmma_f32_16x16x4_f32(
                /*neg_a=*/false, a, /*neg_b=*/false, b,
                /*c_mod=*/(short)0, c, /*reuse_a=*/false, /*reuse_b=*/false);

        // C/D layout: VGPR v -> row (mbase + v), col = t*16 + sub.
        if (col < NFEAT) {
            float* p = out + (size_t)mbase * NFEAT + col;
            #pragma unroll
            for (int v = 0; v < 8; ++v) {
                __builtin_nontemporal_store(__builtin_amdgcn_exp2f(c[v]), p);  // v_exp_f32
                p += NFEAT;   // next row, same column
            }
        }
    }
}

extern "C" void kernel_launch(void* const* d_in, const int* in_sizes, int n_in,
                              void* d_out, int out_size, void* d_ws, size_t ws_size,
                              hipStream_t stream)
{
    const float* X = (const float*)d_in[0];   // [BATCH, 4] f32
    const float* E = (const float*)d_in[1];   // [625, 4] f32
    float* out = (float*)d_out;               // [BATCH, 625] f32

    const int mtiles = BATCH / 16;                       // 16384
    const int blocks = mtiles / WAVES_PER_BLOCK;         // 2048
    formula_feature_wmma_kernel<<<blocks, 256, 0, stream>>>(X, E, out);
}